// SoftReordering_66262755442701
// MI455X (gfx1250) — compile-verified
//
#include <hip/hip_runtime.h>

typedef float v2f __attribute__((ext_vector_type(2)));
typedef float v8f __attribute__((ext_vector_type(8)));
typedef unsigned int u32x4 __attribute__((ext_vector_type(4)));
typedef int i32x4 __attribute__((ext_vector_type(4)));
typedef int i32x8 __attribute__((ext_vector_type(8)));

#define B_       16
#define S_       4096
#define E_       256
#define W_       7
#define WE_      1792          // W * E
#define TILE_    128           // tokens per workgroup
#define NTHREADS 256           // 8 wave32s
#define XROWS    (TILE_ + 6)   // 134 window rows
#define XF       (XROWS * E_)  // 34304 floats of x window
#define WF       (WE_ * 8)     // 14336 floats: (WE/2) pairs * 16 lanes
#define GOFF     (XF + WF)
#define SMEM_FLOATS (XF + WF + TILE_ * 8)   // 49664 floats = 198656 bytes

__global__ __launch_bounds__(NTHREADS)
void soft_reorder_kernel(const float* __restrict__ x,
                         const float* __restrict__ weight,
                         const float* __restrict__ bias,
                         float* __restrict__ out)
{
    extern __shared__ float smem[];
    float* Xlds = smem;          // [134][256] window rows (token t, shift j -> row t+j)
    float* Wlds = smem + XF;     // [WE/2][16] : pair p, slot 2*n+(f&1); n==7 slot zero-padded
    float* Glds = smem + GOFF;   // [128][8] gates

    const int tid = threadIdx.x;
    const int b   = blockIdx.x >> 5;           // 32 tiles per batch
    const int t0  = (blockIdx.x & 31) * TILE_;

    // ---------- stage x window rows [t0-3, t0+130] into LDS ----------
    const bool interior = (t0 >= 3) && (t0 + TILE_ + 3 <= S_);
    if (interior) {
        // Tensor Data Mover: one 2D DMA (134 x 256 f32, contiguous rows) issued by wave 0.
        if ((tid >> 5) == 0) {
            const unsigned lds_base = (unsigned)(uintptr_t)Xlds;   // LDS byte offset = addr[31:0]
            const unsigned long long ga =
                (unsigned long long)(uintptr_t)(x + ((size_t)b * S_ + (size_t)(t0 - 3)) * E_);
            u32x4 g0;
            g0[0] = 1u;                                   // count=1, is_restore=0, no gather
            g0[1] = lds_base;                             // lds_addr  [63:32]
            g0[2] = (unsigned)(ga & 0xffffffffu);         // global_addr[31:0]
            g0[3] = (unsigned)((ga >> 32) & 0x01ffffffu)  // global_addr[56:32]
                  | (2u << 30);                           // type = 2 ("image")
            i32x8 g1;
            g1[0] = 0x00020000;                           // workgroup_mask=0, data_size=2 (4B)
            g1[1] = (int)(256u << 16);                    // tensor_dim0[15:0]=256 @ bits[63:48]
            g1[2] = (int)((unsigned)XROWS << 16);         // tensor_dim1[15:0]=134 @ bits[95:80]
            g1[3] = (int)(256u << 16);                    // tile_dim0=256 @ bits[127:112]
            g1[4] = XROWS;                                // tile_dim1=134, tile_dim2=0
            g1[5] = 256;                                  // tensor_dim0_stride lo (elements)
            g1[6] = 0;                                    // stride hi / dim1_stride (unused, 2D)
            g1[7] = 0;
            i32x4 gz4 = {0, 0, 0, 0};
            i32x8 gz8 = {0, 0, 0, 0, 0, 0, 0, 0};
            __builtin_amdgcn_tensor_load_to_lds(g0, g1, gz4, gz4, gz8, 0);
            __builtin_amdgcn_s_wait_tensorcnt(0);
        }
    } else {
        // edge tiles: manual staging with zero padding outside [0, S)
        const float4* xg = (const float4*)(x + (size_t)b * S_ * E_);
        float4* xl = (float4*)Xlds;
        for (int i = tid; i < XF / 4; i += NTHREADS) {
            const int r = i >> 6;          // row (64 float4 per row)
            const int g = t0 - 3 + r;      // global token index
            float4 v = make_float4(0.f, 0.f, 0.f, 0.f);
            if (g >= 0 && g < S_) v = xg[(size_t)g * 64 + (i & 63)];
            xl[i] = v;
        }
    }
    // ---------- stage weight repacked for B-operand pair loads ----------
    for (int idx = tid; idx < WF; idx += NTHREADS) {
        const int rem  = idx & 15;
        const int pair = idx >> 4;
        const int n    = rem >> 1;
        const int f    = (pair << 1) | (rem & 1);
        Wlds[idx] = (n < W_) ? weight[n * WE_ + f] : 0.0f;
    }
    __syncthreads();

    // ---------- GEMM phase: logits[16 tok x 16 gates] per wave via WMMA f32 16x16x4 ----------
    const int wave = tid >> 5;
    const int lane = tid & 31;
    const int half = lane >> 4;   // 0: K={0,1}, 1: K={2,3}
    const int lp   = lane & 15;   // A: M row, B/C: N column
    const int nm   = (lp < W_) ? lp : W_;   // clamp to zero-padded slot

    v8f acc = {};
    for (int j = 0; j < W_; ++j) {
        const float* Ap = Xlds + (wave * 16 + lp + j) * E_ + half * 2;
        const float* Bp = Wlds + j * 2048 + half * 16 + nm * 2;
#pragma unroll 8
        for (int e0 = 0; e0 < E_; e0 += 4) {
            v2f a  = *(const v2f*)(Ap + e0);
            v2f bb = *(const v2f*)(Bp + (e0 >> 1) * 16);
            acc = __builtin_amdgcn_wmma_f32_16x16x4_f32(
                      false, a, false, bb, (short)0, acc, false, false);
        }
    }

    // epilogue: bias + sigmoid, park gates in LDS
    if (lp < W_) {
        const float bv = bias[lp];
#pragma unroll
        for (int r = 0; r < 8; ++r) {
            const float logit = acc[r] + bv;
            const float gate  = 1.0f / (1.0f + __expf(-logit));
            const int tok = wave * 16 + half * 8 + r;   // C layout: VGPR r -> M = r + 8*half
            Glds[tok * 8 + lp] = gate;
        }
    }
    __syncthreads();

    // ---------- blend phase: score[t,e] = sum_k gate[t,k] * win[t,k,e]; out = tanh ----------
    const int e = tid;   // 256 threads == E channels
    float* og = out + ((size_t)b * S_ + t0) * E_ + e;
    for (int t = 0; t < TILE_; ++t) {
        const float4 g03 = *(const float4*)&Glds[t * 8];
        const float4 g47 = *(const float4*)&Glds[t * 8 + 4];
        const float* xr = Xlds + t * E_ + e;
        float s = g03.x * xr[0 * E_]
                + g03.y * xr[1 * E_]
                + g03.z * xr[2 * E_]
                + g03.w * xr[3 * E_]
                + g47.x * xr[4 * E_]
                + g47.y * xr[5 * E_]
                + g47.z * xr[6 * E_];
        og[(size_t)t * E_] = tanhf(s);
    }
}

extern "C" void kernel_launch(void* const* d_in, const int* in_sizes, int n_in,
                              void* d_out, int out_size, void* d_ws, size_t ws_size,
                              hipStream_t stream) {
    (void)in_sizes; (void)n_in; (void)d_ws; (void)ws_size; (void)out_size;
    const float* x      = (const float*)d_in[0];
    const float* weight = (const float*)d_in[1];
    const float* bias   = (const float*)d_in[2];
    float* out = (float*)d_out;

    dim3 grid(B_ * (S_ / TILE_));   // 512 workgroups
    soft_reorder_kernel<<<grid, NTHREADS, SMEM_FLOATS * sizeof(float), stream>>>(
        x, weight, bias, out);
}